// SpatialGraphBuilder_21277267984966
// MI455X (gfx1250) — compile-verified
//
#include <hip/hip_runtime.h>
#include <hip/hip_bf16.h>
#include <stdint.h>

// ---------------------------------------------------------------------------
// SpatialGraphBuilder on MI455X (gfx1250):
//   w[m,n] = d1[m]vs1[n] - d0[m]vs0[n] + vs1[m]d1[n] - vs0[m]d0[n]
//          = (A @ B^T)[m,n],  A,B : [4096, 4]   ->  V_WMMA_F32_16X16X4_F32
// Score matrix never touches HBM: 3 tile passes (max / hist / collect)
// recompute it via WMMA. B panel is staged global->LDS in 16KB chunks using
// the CDNA5 async-LDS path (ASYNCcnt) when available; inner loop is 8
// pipelined ds_load_b64 + WMMA per wave per chunk.
// ---------------------------------------------------------------------------

#define NN        4096
#define NTILE     256          // 4096 / 16 tiles per dimension
#define CHUNKROWS 1024         // B rows staged per LDS chunk (16 KB)
#define HBINS     2048
#define EDGELIM   65536
#define CAP       131072       // 2^17 over-collect buffer (power of two for bitonic)

typedef __attribute__((ext_vector_type(2))) float v2f;
typedef __attribute__((ext_vector_type(8))) float v8f;
typedef __attribute__((ext_vector_type(2))) int   v2i_t;

#if defined(__has_builtin)
#if __has_builtin(__builtin_amdgcn_global_load_async_to_lds_b64)
#define HAVE_ASYNC_LDS 1
#endif
#endif
#ifndef HAVE_ASYNC_LDS
#define HAVE_ASYNC_LDS 0
#endif

#if HAVE_ASYNC_LDS
// clang signature: (v2i addrspace(1)*, v2i addrspace(3)*, imm offset, imm cpol)
typedef __attribute__((address_space(1))) v2i_t* async_gp_t;
typedef __attribute__((address_space(3))) v2i_t* async_lp_t;
#endif

// D-matrix mapping: VGPR r, lanes 0-15 -> M = r ; lanes 16-31 -> M = r + 8.
__device__ __forceinline__ int tile_row(int m0, int r, int half) { return m0 + r + (half << 3); }
__device__ __forceinline__ int tile_col(int n0, int lane)        { return n0 + (lane & 15); }

// Walk all 256 n-tiles for this block's m-tile. B staged via LDS chunks.
// op(acc, m0, n0, lane, half) consumes each 16x16 WMMA result tile.
template <typename Op>
__device__ __forceinline__ void score_tiles(const float* __restrict__ A4,
                                            const float* __restrict__ B4,
                                            Op&& op) {
  __shared__ __align__(16) float sB[CHUNKROWS * 4];   // 16 KB
  const int m0   = (int)blockIdx.x << 4;
  const int lane = (int)threadIdx.x & 31;             // wave32
  const int wave = (int)threadIdx.x >> 5;             // 8 waves / block
  const int half = lane >> 4;                         // K={0,1} vs K={2,3}
  const int r    = lane & 15;

  // A operand is fixed for the whole block (16x4 f32 A layout).
  v2f a;
  {
    const float* ap = A4 + ((m0 + r) << 2) + (half << 1);
    a.x = ap[0]; a.y = ap[1];
  }

  for (int chunk = 0; chunk < 4; ++chunk) {
    const int rowbase = chunk << 10;                  // 1024 rows per chunk
    __syncthreads();                                  // prior chunk fully consumed

    // ---- stage 16 KB of B4 into LDS ----
#if HAVE_ASYNC_LDS
    {
      const char* gsrc = (const char*)(B4 + (rowbase << 2));
      char* lbase = (char*)sB;
#pragma unroll
      for (int i = 0; i < 8; ++i) {                   // 8 x b64 per thread
        const int off = (((i << 8) + (int)threadIdx.x) << 3);
        __builtin_amdgcn_global_load_async_to_lds_b64(
            (async_gp_t)(gsrc + off), (async_lp_t)(lbase + off), 0, 0);
      }
      asm volatile("s_wait_asynccnt 0x0" ::: "memory");
    }
#else
    {
      const float4* gsrc = (const float4*)(B4 + (rowbase << 2));
      float4* ldst = (float4*)sB;
#pragma unroll
      for (int i = 0; i < 4; ++i)                     // 4 x b128 per thread
        ldst[(i << 8) + (int)threadIdx.x] = gsrc[(i << 8) + (int)threadIdx.x];
    }
#endif
    __syncthreads();

    // Prefetch next chunk of B while this one computes.
    if (chunk < 3)
      __builtin_prefetch(B4 + ((rowbase + CHUNKROWS) << 2) + ((int)threadIdx.x << 2), 0, 3);

    // ---- 8 tiles per wave in this chunk: pipelined ds_load + WMMA ----
#pragma unroll
    for (int t = 0; t < 8; ++t) {
      const int nt = wave + (t << 3);                 // 0..63 local tile
      v2f b;
      const float* bp = sB + (((nt << 4) + r) << 2) + (half << 1);
      b.x = bp[0]; b.y = bp[1];
      v8f c = {};
      // 8 args: (neg_a, A, neg_b, B, c_mod, C, reuse_a, reuse_b)
      v8f acc = __builtin_amdgcn_wmma_f32_16x16x4_f32(false, a, false, b,
                                                      (short)0, c, false, false);
      op(acc, m0, rowbase + (nt << 4), lane, half);
    }
  }
}

// --------------------------- kernel 0: init --------------------------------
__global__ void init_kernel(int* __restrict__ hist, int* __restrict__ maxbits,
                            int* __restrict__ counter, float* __restrict__ thr) {
  int i = blockIdx.x * blockDim.x + threadIdx.x;
  if (i < HBINS) hist[i] = 0;
  if (i == 0) { *maxbits = 0; *counter = 0; *thr = 0.f; }
}

// ------------------- kernel 1: projections + LN -> A4/B4 -------------------
__global__ void prep_kernel(const float* __restrict__ bboxes,
                            const float* __restrict__ Wq,
                            const float* __restrict__ Wk,
                            const float* __restrict__ Wv,
                            const float* __restrict__ gam,
                            const float* __restrict__ bet,
                            float* __restrict__ A4, float* __restrict__ B4) {
  int n = blockIdx.x * blockDim.x + threadIdx.x;
  if (n >= NN) return;
  float bb[8];
#pragma unroll
  for (int f = 0; f < 8; ++f) bb[f] = bboxes[n * 8 + f];

  float qk0 = 0.f, qk1 = 0.f, vs0 = 0.f, vs1 = 0.f;
#pragma unroll
  for (int h = 0; h < 8; ++h) {
    float q0 = 0.f, q1 = 0.f, k0 = 0.f, k1 = 0.f, v0 = 0.f, v1 = 0.f;
#pragma unroll
    for (int f = 0; f < 8; ++f) {
      float x = bb[f];
      q0 = fmaf(x, Wq[(2 * h + 0) * 8 + f], q0);
      q1 = fmaf(x, Wq[(2 * h + 1) * 8 + f], q1);
      k0 = fmaf(x, Wk[(2 * h + 0) * 8 + f], k0);
      k1 = fmaf(x, Wk[(2 * h + 1) * 8 + f], k1);
      v0 = fmaf(x, Wv[(2 * h + 0) * 8 + f], v0);
      v1 = fmaf(x, Wv[(2 * h + 1) * 8 + f], v1);
    }
    qk0 = fmaf(q0, k0, qk0);
    qk1 = fmaf(q1, k1, qk1);
    vs0 += v0; vs1 += v1;
  }
  // LayerNorm over the 2-vector (biased variance, eps = 1e-5)
  float mu  = 0.5f * (qk0 + qk1);
  float c0  = qk0 - mu, c1 = qk1 - mu;
  float var = 0.5f * (c0 * c0 + c1 * c1);
  float rs  = rsqrtf(var + 1e-5f);
  float d0  = c0 * rs * gam[0] + bet[0];
  float d1  = c1 * rs * gam[1] + bet[1];

  A4[n * 4 + 0] = d1;  A4[n * 4 + 1] = -d0;
  A4[n * 4 + 2] = vs1; A4[n * 4 + 3] = -vs0;
  B4[n * 4 + 0] = vs1; B4[n * 4 + 1] = vs0;
  B4[n * 4 + 2] = d1;  B4[n * 4 + 3] = d0;
}

// ---------------- kernel 2: pass over tiles -> max positive w --------------
__global__ void pass_max(const float* __restrict__ A4, const float* __restrict__ B4,
                         int* __restrict__ maxbits) {
  float lmax = 0.f;
  score_tiles(A4, B4, [&](const v8f& acc, int, int, int, int) {
#pragma unroll
    for (int r = 0; r < 8; ++r) lmax = fmaxf(lmax, acc[r]);
  });
  if (lmax > 0.f) atomicMax(maxbits, __float_as_int(lmax));  // positive floats order as ints
}

// ---------------- kernel 3: pass over tiles -> value histogram -------------
__global__ void pass_hist(const float* __restrict__ A4, const float* __restrict__ B4,
                          const int* __restrict__ maxbits, int* __restrict__ ghist) {
  __shared__ int sh[HBINS];
  for (int i = threadIdx.x; i < HBINS; i += blockDim.x) sh[i] = 0;

  const float maxw  = __int_as_float(*maxbits);
  const float scale = (maxw > 0.f) ? ((float)HBINS / maxw) : 0.f;

  // score_tiles begins with __syncthreads(), covering the sh[] zeroing.
  score_tiles(A4, B4, [&](const v8f& acc, int, int, int, int) {
#pragma unroll
    for (int r = 0; r < 8; ++r) {
      float v = acc[r];
      if (v > 0.f) {
        int b = (int)(v * scale);
        b = (b < HBINS - 1) ? b : (HBINS - 1);
        atomicAdd(&sh[b], 1);
      }
    }
  });

  __syncthreads();
  for (int i = threadIdx.x; i < HBINS; i += blockDim.x)
    if (sh[i]) atomicAdd(&ghist[i], sh[i]);
}

// ---------------- kernel 4: scan histogram -> threshold --------------------
__global__ void thresh_kernel(const int* __restrict__ hist,
                              const int* __restrict__ maxbits,
                              float* __restrict__ thr) {
  if (threadIdx.x != 0 || blockIdx.x != 0) return;
  float maxw = __int_as_float(*maxbits);
  float t = 0.f;
  if (maxw > 0.f) {
    long total = 0;
    for (int b = HBINS - 1; b >= 0; --b) {
      total += hist[b];
      if (total >= (long)EDGELIM) { t = (float)b * (maxw / (float)HBINS); break; }
    }
  }
  *thr = t;
}

// ---------------- kernel 5: pass over tiles -> collect candidates ----------
__global__ void pass_collect(const float* __restrict__ A4, const float* __restrict__ B4,
                             const float* __restrict__ thr, int* __restrict__ counter,
                             float* __restrict__ vals, int* __restrict__ idxs) {
  const float t = *thr;
  score_tiles(A4, B4, [&](const v8f& acc, int m0, int n0, int lane, int half) {
#pragma unroll
    for (int r = 0; r < 8; ++r) {
      float v = acc[r];
      if (v > 0.f && v >= t) {
        int p = atomicAdd(counter, 1);
        if (p < CAP) {
          vals[p] = v;
          idxs[p] = tile_row(m0, r, half) * NN + tile_col(n0, lane);
        }
      }
    }
  });
}

// ---------------- kernel 6: pad unused slots with -inf ---------------------
__global__ void pad_kernel(float* __restrict__ vals, int* __restrict__ idxs,
                           const int* __restrict__ counter) {
  int i = blockIdx.x * blockDim.x + threadIdx.x;
  int c = *counter; c = (c < CAP) ? c : CAP;
  if (i >= c && i < CAP) { vals[i] = -__builtin_inff(); idxs[i] = 0x7FFFFFFF; }
}

// ---------------- kernel 7: bitonic sort step (desc val, asc idx) ----------
__global__ void bitonic_kernel(float* __restrict__ vals, int* __restrict__ idxs,
                               unsigned j, unsigned k) {
  unsigned i = blockIdx.x * blockDim.x + threadIdx.x;
  unsigned l = i ^ j;
  if (l > i) {
    float vi = vals[i], vl = vals[l];
    int   ii = idxs[i], il = idxs[l];
    // "i belongs after l" in descending order (value desc, index asc on ties)
    bool iAfter = (vi < vl) || (vi == vl && ii > il);
    bool desc   = ((i & k) == 0);
    if (desc == iAfter) {
      vals[i] = vl; vals[l] = vi;
      idxs[i] = il; idxs[l] = ii;
    }
  }
}

// ---------------- kernel 8: emit edge_index / vals / edge_attr -------------
__global__ void emit_kernel(const float* __restrict__ vals, const int* __restrict__ idxs,
                            const float* __restrict__ rel, const float* __restrict__ Wa,
                            const float* __restrict__ ba, float* __restrict__ out) {
  int e = blockIdx.x * blockDim.x + threadIdx.x;
  if (e >= EDGELIM) return;
  int idx  = idxs[e];
  bool ok  = (unsigned)idx < (unsigned)NN * (unsigned)NN;
  int i    = ok ? (idx >> 12) : 0;
  int j    = ok ? (idx & (NN - 1)) : 0;

  int* oi = (int*)out;                 // edge_index int32 [2, EDGELIM]
  oi[e]           = i;
  oi[EDGELIM + e] = j;
  out[2 * EDGELIM + e] = vals[e];      // top-k values

  float r6[6];
  const float* rp = rel + (size_t)(ok ? idx : 0) * 6;
#pragma unroll
  for (int r = 0; r < 6; ++r) r6[r] = ok ? rp[r] : 0.f;

  float* ap = out + 3 * EDGELIM + (size_t)e * 8;   // edge_attr [EDGELIM, 8]
#pragma unroll
  for (int d = 0; d < 8; ++d) {
    float acc = ba[d];
#pragma unroll
    for (int r = 0; r < 6; ++r) acc = fmaf(r6[r], Wa[d * 6 + r], acc);
    ap[d] = acc;
  }
}

// ---------------------------------------------------------------------------
extern "C" void kernel_launch(void* const* d_in, const int* in_sizes, int n_in,
                              void* d_out, int out_size, void* d_ws, size_t ws_size,
                              hipStream_t stream) {
  const float* bboxes = (const float*)d_in[0];
  const float* rel    = (const float*)d_in[1];
  const float* Wq     = (const float*)d_in[2];
  const float* Wk     = (const float*)d_in[3];
  const float* Wv     = (const float*)d_in[4];
  const float* gam    = (const float*)d_in[5];
  const float* bet    = (const float*)d_in[6];
  const float* Wa     = (const float*)d_in[7];
  const float* ba     = (const float*)d_in[8];
  // d_in[9] = edge_limit (compile-time constant 65536 in this problem)

  char* ws = (char*)d_ws;
  float* A4      = (float*)(ws + 0);                 // 4096*4 f32 = 64 KB
  float* B4      = (float*)(ws + 65536);             // 64 KB
  int*   hist    = (int*)  (ws + 131072);            // 2048 bins = 8 KB
  int*   maxbits = (int*)  (ws + 139264);
  int*   counter = (int*)  (ws + 139268);
  float* thr     = (float*)(ws + 139272);
  float* vals    = (float*)(ws + 147456);            // CAP f32 = 512 KB
  int*   idxs    = (int*)  (ws + 147456 + CAP * 4);  // CAP i32 = 512 KB

  init_kernel  <<<(HBINS + 255) / 256, 256, 0, stream>>>(hist, maxbits, counter, thr);
  prep_kernel  <<<NN / 256, 256, 0, stream>>>(bboxes, Wq, Wk, Wv, gam, bet, A4, B4);
  pass_max     <<<NTILE, 256, 0, stream>>>(A4, B4, maxbits);
  pass_hist    <<<NTILE, 256, 0, stream>>>(A4, B4, maxbits, hist);
  thresh_kernel<<<1, 32, 0, stream>>>(hist, maxbits, thr);
  pass_collect <<<NTILE, 256, 0, stream>>>(A4, B4, thr, counter, vals, idxs);
  pad_kernel   <<<CAP / 256, 256, 0, stream>>>(vals, idxs, counter);

  for (unsigned k = 2; k <= CAP; k <<= 1)
    for (unsigned j = k >> 1; j > 0; j >>= 1)
      bitonic_kernel<<<CAP / 256, 256, 0, stream>>>(vals, idxs, j, k);

  emit_kernel  <<<EDGELIM / 256, 256, 0, stream>>>(vals, idxs, rel, Wa, ba, (float*)d_out);
}